// tfft_83099027243546
// MI455X (gfx1250) — compile-verified
//
#include <hip/hip_runtime.h>

// Batched 4096-point FFT of real input, one workgroup per row.
// Algorithm: DIF radix-16, three passes, each pass a complex 16x16 DFT-matrix
// multiply implemented with V_WMMA_F32_16X16X4_F32 (K=16 via 4 chained WMMAs),
// twiddles from the reference's own cos/-sin table (rconvs[11,1,:]/iconvs[11,1,:]),
// final base-16 digit-reversal folded into the stage-2 LDS store.
// HBM traffic: 32MB read + 64MB write -> ~4.1us roofline at 23.3 TB/s.

#define NFFT 4096
#define TPB  256

typedef float v2f __attribute__((ext_vector_type(2)));
typedef float v8f __attribute__((ext_vector_type(8)));

static __device__ __forceinline__ v8f wmma4(v2f a, v2f b, v8f c) {
  // D = A(16x4,f32) * B(4x16,f32) + C(16x16,f32)
  return __builtin_amdgcn_wmma_f32_16x16x4_f32(false, a, false, b, (short)0, c,
                                               false, false);
}

__global__ __launch_bounds__(TPB) void fft4096_wmma_kernel(
    const float* __restrict__ x,
    const float* __restrict__ rtab,   //  cos(2*pi*m/4096), m in [0,4096)
    const float* __restrict__ itab,   // -sin(2*pi*m/4096)
    float* __restrict__ outR,
    float* __restrict__ outI)
{
  __shared__ float sR[NFFT];
  __shared__ float sI[NFFT];
  __shared__ float sTr[NFFT];   // twiddle table (stages 0-1) -> output staging (stage 2)
  __shared__ float sTi[NFFT];

  const int tid  = threadIdx.x;
  const int row  = blockIdx.x;
  const int lane = tid & 31;
  const int wave = tid >> 5;
  const int m16  = lane & 15;   // A row index; B/C column index
  const int half = lane >> 4;

  // Coalesced load of input row (imag = 0 implicitly) + twiddle tables.
  for (int k = tid; k < NFFT; k += TPB) {
    sR[k]  = x[(size_t)row * NFFT + k];
    sTr[k] = rtab[k];
    sTi[k] = itab[k];
  }
  __syncthreads();

  // DFT16 matrix D[k][n] = W16^{k n} = rtab[256*((k*n)&15)] + i*itab[...]
  // held in WMMA A-layout: chunk c covers K = 4c..4c+3.
  v2f dr[4], di[4];
#pragma unroll
  for (int c = 0; c < 4; ++c) {
    const int k0 = 4 * c + 2 * half;
    const int i0 = ((m16 * k0) & 15) << 8;
    const int i1 = ((m16 * (k0 + 1)) & 15) << 8;
    dr[c].x = sTr[i0]; dr[c].y = sTr[i1];
    di[c].x = sTi[i0]; di[c].y = sTi[i1];
  }

  // ---------- stage 0: 16-pt DFTs over stride-256 groups; Im(x)=0 -> 8 WMMAs/tile
#pragma unroll
  for (int rep = 0; rep < 2; ++rep) {
    const int t = wave + 8 * rep;          // tile: columns g = 16t..16t+15
    const int g = 16 * t + m16;
    v2f xr[4];
#pragma unroll
    for (int c = 0; c < 4; ++c) {
      const int k0 = 4 * c + 2 * half;
      xr[c].x = sR[k0 * 256 + g];
      xr[c].y = sR[(k0 + 1) * 256 + g];
    }
    v8f yr = {}, yi = {};
#pragma unroll
    for (int c = 0; c < 4; ++c) yr = wmma4(dr[c], xr[c], yr);
#pragma unroll
    for (int c = 0; c < 4; ++c) yi = wmma4(di[c], xr[c], yi);
    // twiddle W_4096^{g*k} then store in place at k*256 + g
#pragma unroll
    for (int v = 0; v < 8; ++v) {
      const int r   = 8 * half + v;
      const int idx = (r * g) & (NFFT - 1);
      const float tr = sTr[idx], ti = sTi[idx];
      const float a = yr[v], b = yi[v];
      sR[r * 256 + g] = a * tr - b * ti;
      sI[r * 256 + g] = a * ti + b * tr;
    }
  }
  __syncthreads();

  // ---------- stage 1: in each 256-block, 16-pt DFTs over stride-16 groups
#pragma unroll
  for (int rep = 0; rep < 2; ++rep) {
    const int t = wave + 8 * rep;          // block index
    const int base = 256 * t;
    v2f xr[4], xi[4];
#pragma unroll
    for (int c = 0; c < 4; ++c) {
      const int k0 = 4 * c + 2 * half;
      xr[c].x = sR[base + k0 * 16 + m16];
      xr[c].y = sR[base + (k0 + 1) * 16 + m16];
      xi[c].x = sI[base + k0 * 16 + m16];
      xi[c].y = sI[base + (k0 + 1) * 16 + m16];
    }
    v8f p = {}, q = {}, wi = {};
#pragma unroll
    for (int c = 0; c < 4; ++c) p = wmma4(dr[c], xr[c], p);
#pragma unroll
    for (int c = 0; c < 4; ++c) q = wmma4(di[c], xi[c], q);
    v8f wr = p - q;                         // Re = Dr*Xr - Di*Xi
#pragma unroll
    for (int c = 0; c < 4; ++c) wi = wmma4(dr[c], xi[c], wi);
#pragma unroll
    for (int c = 0; c < 4; ++c) wi = wmma4(di[c], xr[c], wi);   // Im = Dr*Xi + Di*Xr
    // twiddle W_256^{j*k} = tab[(16*j*k)&4095], store at base + k*16 + j
#pragma unroll
    for (int v = 0; v < 8; ++v) {
      const int r   = 8 * half + v;
      const int idx = (16 * r * m16) & (NFFT - 1);
      const float tr = sTr[idx], ti = sTi[idx];
      const float a = wr[v], b = wi[v];
      sR[base + r * 16 + m16] = a * tr - b * ti;
      sI[base + r * 16 + m16] = a * ti + b * tr;
    }
  }
  __syncthreads();

  // ---------- stage 2: 16-pt DFTs on contiguous 16-blocks; digit-reversed store
  // into staging (reuses twiddle-table LDS; stage 2 needs no twiddles).
#pragma unroll
  for (int rep = 0; rep < 2; ++rep) {
    const int t = wave + 8 * rep;
    v2f xr[4], xi[4];
#pragma unroll
    for (int c = 0; c < 4; ++c) {
      const int k0 = 4 * c + 2 * half;
      const int p0 = (16 * t + m16) * 16;  // group g = 16t + col, elements contiguous
      xr[c].x = sR[p0 + k0];
      xr[c].y = sR[p0 + k0 + 1];
      xi[c].x = sI[p0 + k0];
      xi[c].y = sI[p0 + k0 + 1];
    }
    v8f p = {}, q = {}, wi = {};
#pragma unroll
    for (int c = 0; c < 4; ++c) p = wmma4(dr[c], xr[c], p);
#pragma unroll
    for (int c = 0; c < 4; ++c) q = wmma4(di[c], xi[c], q);
    v8f wr = p - q;
#pragma unroll
    for (int c = 0; c < 4; ++c) wi = wmma4(dr[c], xi[c], wi);
#pragma unroll
    for (int c = 0; c < 4; ++c) wi = wmma4(di[c], xr[c], wi);
    // final index for value (row r, group g=16t+col): 256*r + 16*(g%16) + g/16
#pragma unroll
    for (int v = 0; v < 8; ++v) {
      const int r = 8 * half + v;
      sTr[256 * r + 16 * m16 + t] = wr[v];
      sTi[256 * r + 16 * m16 + t] = wi[v];
    }
  }
  __syncthreads();

  // Coalesced store of the fully ordered spectrum.
  for (int k = tid; k < NFFT; k += TPB) {
    outR[(size_t)row * NFFT + k] = sTr[k];
    outI[(size_t)row * NFFT + k] = sTi[k];
  }
}

extern "C" void kernel_launch(void* const* d_in, const int* in_sizes, int n_in,
                              void* d_out, int out_size, void* d_ws, size_t ws_size,
                              hipStream_t stream) {
  // inputs: x (B*4096 f32), perm (unused), rconvs (12*2*4096), iconvs (12*2*4096)
  const float* x     = (const float*)d_in[0];
  const float* rconv = (const float*)d_in[2];
  const float* iconv = (const float*)d_in[3];
  const int batch = in_sizes[0] / NFFT;
  // rconvs[11,1,m] == cos(2*pi*m/4096), iconvs[11,1,m] == -sin(2*pi*m/4096), all m
  const float* rtab = rconv + (size_t)(11 * 2 + 1) * NFFT;
  const float* itab = iconv + (size_t)(11 * 2 + 1) * NFFT;
  float* outR = (float*)d_out;
  float* outI = outR + (size_t)batch * NFFT;
  fft4096_wmma_kernel<<<batch, TPB, 0, stream>>>(x, rtab, itab, outR, outI);
}